// DeepConvexFlow_11373073400418
// MI455X (gfx1250) — compile-verified
//
#include <hip/hip_runtime.h>
#include <hip/hip_bf16.h>
#include <math.h>

// ---------------------------------------------------------------------------
// DeepConvexFlow: f = grad F, logdet(Hess F) per sample.
// N=8192 samples, D=64 input dim, H=128 hidden.
// Hess ICNN = sum_k J_k^T D_k J_k ; H_total = sp(w1)*Hess + sp(w0)*I (SPD).
// Compute-bound (~64 GFLOP vs ~4MB HBM) -> f32 WMMA 16x16x4 for all GEMMs.
// ---------------------------------------------------------------------------

#define NN 8192
#define DD 64
#define HH 128

// workspace layout (floats): row-major and transposed copies of sp(Wz*)
#define WS_U1   0            // sp(Wz1)          : 128*128
#define WS_U2   16384        // sp(Wz2)          : 128*128
#define WS_U1T  32768        // sp(Wz1)^T        : 128*128
#define WS_U2T  49152        // sp(Wz2)^T        : 128*128
#define WS_U3   65536        // sp(wz_out)       : 128
#define WS_SCAL 65664        // [sp(w0), sp(w1)]

// LDS layout (floats), per 1-sample workgroup
#define JL 72                // padded row stride for J1/J2 (bank-conflict free)
#define HL 65                // padded row stride for H
#define OFF_J1  0
#define OFF_J2  (OFF_J1 + HH*JL)          // 9216
#define OFF_H   (OFF_J2 + HH*JL)          // 18432
#define OFF_X   (OFF_H  + DD*HL)          // 22592
#define OFF_V   (OFF_X  + DD)             // 22656 ; 10 vectors of 128
#define SMEM_FLOATS (OFF_V + 10*HH)
#define SMEM_BYTES  (SMEM_FLOATS*4)       // ~95.5 KB  (< 320KB/WGP)

typedef __attribute__((ext_vector_type(2))) float v2f;
typedef __attribute__((ext_vector_type(8))) float v8f;

static __device__ __forceinline__ float sigm(float x) {
    return 1.0f / (1.0f + expf(-x));
}
static __device__ __forceinline__ float splus(float x) {
    return fmaxf(x, 0.0f) + log1pf(expf(-fabsf(x)));
}
static __device__ __forceinline__ v8f wmma4(v2f a, v2f b, v8f c) {
    // D(16x16,f32) = A(16x4,f32) * B(4x16,f32) + C
    return __builtin_amdgcn_wmma_f32_16x16x4_f32(
        false, a, false, b, (short)0, c, false, false);
}

// ---------------------------------------------------------------------------
// Kernel 0: precompute softplus'd weights (+ transposed copies) once.
// ---------------------------------------------------------------------------
__global__ void dcf_precompute(const float* __restrict__ Wz1,
                               const float* __restrict__ Wz2,
                               const float* __restrict__ wz_out,
                               const float* __restrict__ w0,
                               const float* __restrict__ w1,
                               float* __restrict__ ws) {
    int i = blockIdx.x * blockDim.x + threadIdx.x;
    if (i < HH * HH) {
        float u1 = splus(Wz1[i]);
        float u2 = splus(Wz2[i]);
        ws[WS_U1 + i] = u1;
        ws[WS_U2 + i] = u2;
        int r = i >> 7, c = i & (HH - 1);
        ws[WS_U1T + c * HH + r] = u1;
        ws[WS_U2T + c * HH + r] = u2;
    }
    if (i < HH) ws[WS_U3 + i] = splus(wz_out[i]);
    if (i == 0) { ws[WS_SCAL + 0] = splus(w0[0]); ws[WS_SCAL + 1] = splus(w1[0]); }
}

// ---------------------------------------------------------------------------
// Kernel 1: one workgroup (128 thr = 4 waves) per sample.
// ---------------------------------------------------------------------------
__global__ __launch_bounds__(128)
void dcf_main(const float* __restrict__ x_in,
              const float* __restrict__ logdet_in,
              const float* __restrict__ Wx0, const float* __restrict__ b0,
              const float* __restrict__ Wx1, const float* __restrict__ b1,
              const float* __restrict__ Wx2, const float* __restrict__ b2,
              const float* __restrict__ wx_out,
              const float* __restrict__ ws,
              float* __restrict__ f_out,
              float* __restrict__ ld_out) {
    extern __shared__ float sm[];
    float* sJ1 = sm + OFF_J1;   // J1: [HH][JL]
    float* sJ2 = sm + OFF_J2;   // J2: [HH][JL]
    float* sH  = sm + OFF_H;    // H : [DD][HL]
    float* sx  = sm + OFF_X;    // x : [DD]
    float* s0v = sm + OFF_V + 0*HH;   // sigma(a0)
    float* z0v = sm + OFF_V + 1*HH;
    float* s1v = sm + OFF_V + 2*HH;
    float* z1v = sm + OFF_V + 3*HH;
    float* d0v = sm + OFF_V + 4*HH;   // g0*sigma(a0)
    float* d1v = sm + OFF_V + 5*HH;
    float* d2v = sm + OFF_V + 6*HH;
    float* d0w = sm + OFF_V + 7*HH;   // g0*sigma'(a0)  (diag D0)
    float* d1w = sm + OFF_V + 8*HH;
    float* d2w = sm + OFF_V + 9*HH;

    const int tid  = threadIdx.x;
    const int lane = tid & 31;
    const int wv   = tid >> 5;        // 0..3
    const int lr   = lane & 15;       // tile row (A) / tile col (B,C,D)
    const int lh   = lane >> 4;       // half selector

    const int n = blockIdx.x;         // sample index
    const float* U1  = ws + WS_U1;
    const float* U2  = ws + WS_U2;
    const float* U1T = ws + WS_U1T;
    const float* U2T = ws + WS_U2T;
    const float* u3  = ws + WS_U3;
    const float spw0 = ws[WS_SCAL + 0];
    const float spw1 = ws[WS_SCAL + 1];

    // ---- load x ----
    if (tid < DD) sx[tid] = x_in[(size_t)n * DD + tid];
    __syncthreads();

    // ---- forward (VALU matvecs; contiguous row picked per layout) ----
    const int h = tid;                // 0..127 == H
    {
        float a0 = b0[h];
        #pragma unroll 8
        for (int d = 0; d < DD; ++d) a0 = fmaf(sx[d], Wx0[d * HH + h], a0);
        s0v[h] = sigm(a0);
        z0v[h] = splus(a0);
    }
    __syncthreads();
    {
        const float* row = U1T + h * HH;   // col h of U1 == row h of U1T
        float a1 = b1[h];
        #pragma unroll 8
        for (int j = 0; j < HH; ++j) a1 = fmaf(z0v[j], row[j], a1);
        #pragma unroll 8
        for (int d = 0; d < DD; ++d) a1 = fmaf(sx[d], Wx1[d * HH + h], a1);
        s1v[h] = sigm(a1);
        z1v[h] = splus(a1);
    }
    __syncthreads();
    {
        const float* row = U2T + h * HH;
        float a2 = b2[h];
        #pragma unroll 8
        for (int j = 0; j < HH; ++j) a2 = fmaf(z1v[j], row[j], a2);
        #pragma unroll 8
        for (int d = 0; d < DD; ++d) a2 = fmaf(sx[d], Wx2[d * HH + h], a2);
        float s2 = sigm(a2);
        float g2 = u3[h];
        d2v[h] = g2 * s2;
        d2w[h] = g2 * s2 * (1.0f - s2);
    }
    __syncthreads();
    {   // g1 = U2 * d2  (row h of U2, contiguous)
        const float* row = U2 + h * HH;
        float g1 = 0.0f;
        #pragma unroll 8
        for (int j = 0; j < HH; ++j) g1 = fmaf(row[j], d2v[j], g1);
        float s1 = s1v[h];
        d1v[h] = g1 * s1;
        d1w[h] = g1 * s1 * (1.0f - s1);
    }
    __syncthreads();
    {   // g0 = U1 * d1
        const float* row = U1 + h * HH;
        float g0 = 0.0f;
        #pragma unroll 8
        for (int j = 0; j < HH; ++j) g0 = fmaf(row[j], d1v[j], g0);
        float s0 = s0v[h];
        d0v[h] = g0 * s0;
        d0w[h] = g0 * s0 * (1.0f - s0);
    }
    __syncthreads();

    // ---- f = sp(w1)*(W0 d0 + W1 d1 + W2 d2 + wx_out) + sp(w0)*x ----
    if (tid < DD) {
        const int d = tid;
        float g = wx_out[d];
        #pragma unroll 8
        for (int j = 0; j < HH; ++j) {
            g = fmaf(Wx0[d * HH + j], d0v[j], g);
            g = fmaf(Wx1[d * HH + j], d1v[j], g);
            g = fmaf(Wx2[d * HH + j], d2v[j], g);
        }
        f_out[(size_t)n * DD + d] = spw1 * g + spw0 * sx[d];
    }

    // =======================================================================
    // WMMA phase.  f32 16x16x4 lane layouts (ISA 7.12.2):
    //   A(16x4):  lane = m + 16*(k>>1), vgpr = k&1  -> per-lane pair k,k+1
    //   B(4x16):  lane = n + 16*(k>>1), vgpr = k&1
    //   C/D:      M = r + 8*(lane>>4), N = lane&15
    // A pairs are contiguous via the transposed U copies -> b64 loads.
    // =======================================================================

    // ---- J1 = U1^T * (diag(s0) * W0^T) + W1^T   (128 x 64) ----
    for (int t = wv; t < 32; t += 4) {
        const int mb = (t >> 2) * 16;     // row-block in H
        const int nb = (t & 3) * 16;      // col-block in D
        const float* Arow = U1T + (mb + lr) * HH;   // A[m,k] = U1T[m][k]
        const float* Brow = Wx0 + (nb + lr) * HH;   // W0^T[k,n] = Wx0[n][k]
        v8f acc;
        #pragma unroll
        for (int r = 0; r < 8; ++r)
            acc[r] = Wx1[(nb + lr) * HH + (mb + r + 8 * lh)];
        #pragma unroll 8
        for (int k0 = 0; k0 < HH; k0 += 4) {
            const int ka = k0 + 2 * lh;
            v2f a = *(const v2f*)(Arow + ka);
            v2f b = *(const v2f*)(Brow + ka) * *(const v2f*)(s0v + ka);
            acc = wmma4(a, b, acc);
        }
        #pragma unroll
        for (int r = 0; r < 8; ++r)
            sJ1[(mb + r + 8 * lh) * JL + nb + lr] = acc[r];
    }
    __syncthreads();

    // ---- J2 = U2^T * (diag(s1) * J1) + W2^T   (128 x 64) ----
    for (int t = wv; t < 32; t += 4) {
        const int mb = (t >> 2) * 16;
        const int nb = (t & 3) * 16;
        const float* Arow = U2T + (mb + lr) * HH;
        const float* Bcol = sJ1 + nb + lr;          // B[k,n] = s1[k]*J1[k][n]
        v8f acc;
        #pragma unroll
        for (int r = 0; r < 8; ++r)
            acc[r] = Wx2[(nb + lr) * HH + (mb + r + 8 * lh)];
        #pragma unroll 8
        for (int k0 = 0; k0 < HH; k0 += 4) {
            const int ka = k0 + 2 * lh;
            v2f a = *(const v2f*)(Arow + ka);
            v2f s = *(const v2f*)(s1v + ka);
            v2f b;
            b.x = s.x * Bcol[(ka    ) * JL];
            b.y = s.y * Bcol[(ka + 1) * JL];
            acc = wmma4(a, b, acc);
        }
        #pragma unroll
        for (int r = 0; r < 8; ++r)
            sJ2[(mb + r + 8 * lh) * JL + nb + lr] = acc[r];
    }
    __syncthreads();

    // ---- H = sp(w1) * sum_k J_k^T D_k J_k + sp(w0) * I   (64 x 64) ----
    for (int t = wv; t < 16; t += 4) {
        const int mb = (t >> 2) * 16;
        const int nb = (t & 3) * 16;
        v8f acc;
        #pragma unroll
        for (int r = 0; r < 8; ++r) acc[r] = 0.0f;

        {   // term 0: J0 = W0^T ; A[m,k] = D0[k]*Wx0[m][k], B[k,n] = Wx0[n][k]
            const float* Arow = Wx0 + (mb + lr) * HH;
            const float* Brow = Wx0 + (nb + lr) * HH;
            #pragma unroll 8
            for (int k0 = 0; k0 < HH; k0 += 4) {
                const int ka = k0 + 2 * lh;
                v2f a = *(const v2f*)(Arow + ka) * *(const v2f*)(d0w + ka);
                v2f b = *(const v2f*)(Brow + ka);
                acc = wmma4(a, b, acc);
            }
        }
        {   // term 1: J1
            const float* Acol = sJ1 + mb + lr;
            const float* Bcol = sJ1 + nb + lr;
            #pragma unroll 8
            for (int k0 = 0; k0 < HH; k0 += 4) {
                const int ka = k0 + 2 * lh;
                v2f w = *(const v2f*)(d1w + ka);
                v2f a, b;
                a.x = w.x * Acol[(ka    ) * JL];
                a.y = w.y * Acol[(ka + 1) * JL];
                b.x = Bcol[(ka    ) * JL];
                b.y = Bcol[(ka + 1) * JL];
                acc = wmma4(a, b, acc);
            }
        }
        {   // term 2: J2
            const float* Acol = sJ2 + mb + lr;
            const float* Bcol = sJ2 + nb + lr;
            #pragma unroll 8
            for (int k0 = 0; k0 < HH; k0 += 4) {
                const int ka = k0 + 2 * lh;
                v2f w = *(const v2f*)(d2w + ka);
                v2f a, b;
                a.x = w.x * Acol[(ka    ) * JL];
                a.y = w.y * Acol[(ka + 1) * JL];
                b.x = Bcol[(ka    ) * JL];
                b.y = Bcol[(ka + 1) * JL];
                acc = wmma4(a, b, acc);
            }
        }

        #pragma unroll
        for (int r = 0; r < 8; ++r) {
            const int gm = mb + r + 8 * lh;
            const int gn = nb + lr;
            sH[gm * HL + gn] = spw1 * acc[r] + (gm == gn ? spw0 : 0.0f);
        }
    }
    __syncthreads();

    // ---- logdet via Gaussian elimination (H is SPD, no pivoting) ----
    float logsum = 0.0f;
    for (int j = 0; j < DD; ++j) {
        __syncthreads();
        const float piv = sH[j * HL + j];
        if (tid == 0) logsum += logf(fabsf(piv));
        if (tid > j && tid < DD) {
            const float m = sH[tid * HL + j] / piv;
            #pragma unroll 4
            for (int c = j + 1; c < DD; ++c)
                sH[tid * HL + c] = fmaf(-m, sH[j * HL + c], sH[tid * HL + c]);
        }
    }
    if (tid == 0) ld_out[n] = logdet_in[n] + logsum;
}

// ---------------------------------------------------------------------------
extern "C" void kernel_launch(void* const* d_in, const int* in_sizes, int n_in,
                              void* d_out, int out_size, void* d_ws, size_t ws_size,
                              hipStream_t stream) {
    const float* x      = (const float*)d_in[0];
    const float* logdet = (const float*)d_in[1];
    const float* w0     = (const float*)d_in[2];
    const float* w1     = (const float*)d_in[3];
    const float* Wx0    = (const float*)d_in[4];
    const float* b0     = (const float*)d_in[5];
    const float* Wz1    = (const float*)d_in[6];
    const float* Wx1    = (const float*)d_in[7];
    const float* b1     = (const float*)d_in[8];
    const float* Wz2    = (const float*)d_in[9];
    const float* Wx2    = (const float*)d_in[10];
    const float* b2     = (const float*)d_in[11];
    const float* wz_out = (const float*)d_in[12];
    const float* wx_out = (const float*)d_in[13];
    // d_in[14] = b_out: does not affect grad or Hessian.

    float* ws = (float*)d_ws;
    float* f  = (float*)d_out;                 // (N, D)
    float* ld = f + (size_t)NN * DD;           // (N,)

    dcf_precompute<<<(HH * HH + 255) / 256, 256, 0, stream>>>(
        Wz1, Wz2, wz_out, w0, w1, ws);

    dcf_main<<<NN, 128, SMEM_BYTES, stream>>>(
        x, logdet, Wx0, b0, Wx1, b1, Wx2, b2, wx_out, ws, f, ld);
}